// SimpleGAT_19911468384539
// MI455X (gfx1250) — compile-verified
//
#include <hip/hip_runtime.h>
#include <math.h>

#define N_NODES 10000
#define N_EDGES 320000
#define DIM_IN  128
#define DIM_H   32
#define HEADS   8
#define H1      256    // HEADS * DIM_H
#define DIM_OUT 64
#define NEG_SLOPE 0.2f

typedef __attribute__((ext_vector_type(2))) float v2f;
typedef __attribute__((ext_vector_type(8))) float v8f;

// ---------------------------------------------------------------------------
// Utility: float atomic max via int/uint ordering trick (works for all signs,
// initialized with -inf).
// ---------------------------------------------------------------------------
__device__ __forceinline__ void atomicMaxF32(float* addr, float value) {
    if (value >= 0.0f) {
        atomicMax((int*)addr, __float_as_int(value));
    } else {
        atomicMin((unsigned int*)addr, __float_as_uint(value));
    }
}

__global__ void fill_f32(float* __restrict__ p, float v, int n) {
    int i = blockIdx.x * blockDim.x + threadIdx.x;
    if (i < n) p[i] = v;
}

// ---------------------------------------------------------------------------
// GEMM: C[M,Nc] = A[M,K] @ B[K,Nc] + bias[Nc]   (all fp32, row-major)
// One wave per 16x16 output tile, V_WMMA_F32_16X16X4_F32, K stepped by 4.
// ---------------------------------------------------------------------------
__global__ void gemm_wmma_f32(const float* __restrict__ A,
                              const float* __restrict__ B,
                              const float* __restrict__ bias,
                              float* __restrict__ C,
                              int M, int K, int Nc) {
    const int lane   = threadIdx.x & 31;
    const int waveId = threadIdx.x >> 5;
    const int tile   = blockIdx.x * (blockDim.x >> 5) + waveId;
    const int tilesN = Nc >> 4;
    const int tm = tile / tilesN;
    const int tn = tile - tm * tilesN;
    if (tm * 16 >= M) return;                 // wave-uniform exit (EXEC stays all-1)

    const int m0 = tm << 4, n0 = tn << 4;
    const int lrow  = lane & 15;              // M index for A, N index for B
    const int khalf = (lane >> 4) << 1;       // lanes 16-31 hold K+2,K+3

    v8f acc = {};
    for (int k0 = 0; k0 < K; k0 += 4) {
        v2f a, b;
        const float* ap = A + (size_t)(m0 + lrow) * K + (k0 + khalf);
        a.x = ap[0];
        a.y = ap[1];
        const float* bp = B + (size_t)(k0 + khalf) * Nc + (n0 + lrow);
        b.x = bp[0];
        b.y = bp[Nc];
        acc = __builtin_amdgcn_wmma_f32_16x16x4_f32(
            false, a, false, b, (short)0, acc, false, false);
    }

    const int mAdd = (lane >> 4) << 3;        // lanes 16-31 write rows 8..15
    const float bv = bias[n0 + lrow];
#pragma unroll
    for (int r = 0; r < 8; ++r) {
        C[(size_t)(m0 + r + mAdd) * Nc + (n0 + lrow)] = acc[r] + bv;
    }
}

// ---------------------------------------------------------------------------
// Layer 1 edge passes (8 heads x 32 dims)
// ---------------------------------------------------------------------------
__global__ void edge_logits1(const float* __restrict__ xl, const float* __restrict__ xr,
                             const int* __restrict__ src, const int* __restrict__ dst,
                             const float* __restrict__ att,
                             float* __restrict__ logits, float* __restrict__ mmax) {
    int t = blockIdx.x * blockDim.x + threadIdx.x;   // (edge, head)
    if (t >= N_EDGES * HEADS) return;
    int e = t >> 3, h = t & 7;
    int s = src[e], d = dst[e];
    const float* pl = xl + (size_t)s * H1 + h * DIM_H;
    const float* pr = xr + (size_t)d * H1 + h * DIM_H;
    const float* pa = att + h * DIM_H;
    float acc = 0.0f;
#pragma unroll 8
    for (int i = 0; i < DIM_H; ++i) {
        float v = pl[i] + pr[i];
        v = (v > 0.0f) ? v : NEG_SLOPE * v;
        acc += v * pa[i];
    }
    logits[t] = acc;
    atomicMaxF32(&mmax[d * HEADS + h], acc);
}

__global__ void edge_exp1(const int* __restrict__ dst,
                          float* __restrict__ logits,            // in: logit, out: exp
                          const float* __restrict__ mmax,
                          float* __restrict__ denom) {
    int t = blockIdx.x * blockDim.x + threadIdx.x;
    if (t >= N_EDGES * HEADS) return;
    int e = t >> 3, h = t & 7;
    int d = dst[e];
    float ex = expf(logits[t] - mmax[d * HEADS + h]);
    logits[t] = ex;
    atomicAdd(&denom[d * HEADS + h], ex);
}

__global__ void edge_msg1(const float* __restrict__ xl,
                          const int* __restrict__ src, const int* __restrict__ dst,
                          const float* __restrict__ exv, const float* __restrict__ denom,
                          float* __restrict__ out) {
    int e = blockIdx.x;          // one block (256 threads) per edge
    int c = threadIdx.x;         // 0..255
    int h = c >> 5;
    int s = src[e], d = dst[e];
    float alpha = exv[(size_t)e * HEADS + h] / denom[d * HEADS + h];
    atomicAdd(&out[(size_t)d * H1 + c], xl[(size_t)s * H1 + c] * alpha);
}

__global__ void elu_bias(float* __restrict__ h, const float* __restrict__ bias, int n) {
    int i = blockIdx.x * blockDim.x + threadIdx.x;
    if (i >= n) return;
    float v = h[i] + bias[i & (H1 - 1)];
    h[i] = (v > 0.0f) ? v : expm1f(v);
}

// ---------------------------------------------------------------------------
// Layer 2 edge passes (1 head x 64 dims)
// ---------------------------------------------------------------------------
__global__ void edge_logits2(const float* __restrict__ xl, const float* __restrict__ xr,
                             const int* __restrict__ src, const int* __restrict__ dst,
                             const float* __restrict__ att,
                             float* __restrict__ logits, float* __restrict__ mmax) {
    int e = blockIdx.x * blockDim.x + threadIdx.x;
    if (e >= N_EDGES) return;
    int s = src[e], d = dst[e];
    const float* pl = xl + (size_t)s * DIM_OUT;
    const float* pr = xr + (size_t)d * DIM_OUT;
    float acc = 0.0f;
#pragma unroll 8
    for (int i = 0; i < DIM_OUT; ++i) {
        float v = pl[i] + pr[i];
        v = (v > 0.0f) ? v : NEG_SLOPE * v;
        acc += v * att[i];
    }
    logits[e] = acc;
    atomicMaxF32(&mmax[d], acc);
}

__global__ void edge_exp2(const int* __restrict__ dst,
                          float* __restrict__ logits,
                          const float* __restrict__ mmax,
                          float* __restrict__ denom) {
    int e = blockIdx.x * blockDim.x + threadIdx.x;
    if (e >= N_EDGES) return;
    int d = dst[e];
    float ex = expf(logits[e] - mmax[d]);
    logits[e] = ex;
    atomicAdd(&denom[d], ex);
}

__global__ void edge_msg2(const float* __restrict__ xl,
                          const int* __restrict__ src, const int* __restrict__ dst,
                          const float* __restrict__ exv, const float* __restrict__ denom,
                          float* __restrict__ out) {
    int t = blockIdx.x * blockDim.x + threadIdx.x;
    if (t >= N_EDGES * DIM_OUT) return;
    int e = t >> 6, c = t & 63;
    int s = src[e], d = dst[e];
    float alpha = exv[e] / denom[d];
    atomicAdd(&out[(size_t)d * DIM_OUT + c], xl[(size_t)s * DIM_OUT + c] * alpha);
}

// ---------------------------------------------------------------------------
// Row-wise log-softmax over 64 columns; one wave (32 lanes x 2 cols) per row.
// ---------------------------------------------------------------------------
__global__ void log_softmax64(const float* __restrict__ in, const float* __restrict__ bias,
                              float* __restrict__ out, int n) {
    int row  = blockIdx.x * (blockDim.x >> 5) + (threadIdx.x >> 5);
    int lane = threadIdx.x & 31;
    if (row >= n) return;
    float v0 = in[(size_t)row * 64 + lane]      + bias[lane];
    float v1 = in[(size_t)row * 64 + 32 + lane] + bias[32 + lane];
    float mx = fmaxf(v0, v1);
    for (int off = 16; off > 0; off >>= 1) mx = fmaxf(mx, __shfl_xor(mx, off, 32));
    float s = expf(v0 - mx) + expf(v1 - mx);
    for (int off = 16; off > 0; off >>= 1) s += __shfl_xor(s, off, 32);
    float lse = mx + logf(s);
    out[(size_t)row * 64 + lane]      = v0 - lse;
    out[(size_t)row * 64 + 32 + lane] = v1 - lse;
}

// ---------------------------------------------------------------------------
// Launcher
// ---------------------------------------------------------------------------
extern "C" void kernel_launch(void* const* d_in, const int* in_sizes, int n_in,
                              void* d_out, int out_size, void* d_ws, size_t ws_size,
                              hipStream_t stream) {
    const float* x     = (const float*)d_in[0];
    const int*   edge  = (const int*)  d_in[1];
    const float* Wl1   = (const float*)d_in[2];
    const float* bl1   = (const float*)d_in[3];
    const float* Wr1   = (const float*)d_in[4];
    const float* br1   = (const float*)d_in[5];
    const float* att1  = (const float*)d_in[6];
    const float* bias1 = (const float*)d_in[7];
    const float* Wl2   = (const float*)d_in[8];
    const float* bl2   = (const float*)d_in[9];
    const float* Wr2   = (const float*)d_in[10];
    const float* br2   = (const float*)d_in[11];
    const float* att2  = (const float*)d_in[12];
    const float* bias2 = (const float*)d_in[13];

    const int* src = edge;            // edge_index[0]
    const int* dst = edge + N_EDGES;  // edge_index[1]

    float* ws  = (float*)d_ws;
    float* XL1 = ws;  ws += (size_t)N_NODES * H1;       // 2.56M
    float* XR1 = ws;  ws += (size_t)N_NODES * H1;       // 2.56M
    float* LG1 = ws;  ws += (size_t)N_EDGES * HEADS;    // 2.56M (logits -> exp)
    float* M1  = ws;  ws += (size_t)N_NODES * HEADS;
    float* D1  = ws;  ws += (size_t)N_NODES * HEADS;
    float* O1  = ws;  ws += (size_t)N_NODES * H1;       // layer-1 out / h
    float* XL2 = ws;  ws += (size_t)N_NODES * DIM_OUT;
    float* XR2 = ws;  ws += (size_t)N_NODES * DIM_OUT;
    float* LG2 = ws;  ws += (size_t)N_EDGES;
    float* M2  = ws;  ws += (size_t)N_NODES;
    float* D2  = ws;  ws += (size_t)N_NODES;
    float* O2  = ws;  ws += (size_t)N_NODES * DIM_OUT;

    const float NEG_INF = -__builtin_huge_valf();
    auto cdiv = [](int a, int b) { return (a + b - 1) / b; };

    // --- init accumulators ---
    fill_f32<<<cdiv(N_NODES * HEADS, 256), 256, 0, stream>>>(M1, NEG_INF, N_NODES * HEADS);
    fill_f32<<<cdiv(N_NODES * HEADS, 256), 256, 0, stream>>>(D1, 0.0f,    N_NODES * HEADS);
    fill_f32<<<cdiv(N_NODES * H1,    256), 256, 0, stream>>>(O1, 0.0f,    N_NODES * H1);
    fill_f32<<<cdiv(N_NODES,         256), 256, 0, stream>>>(M2, NEG_INF, N_NODES);
    fill_f32<<<cdiv(N_NODES,         256), 256, 0, stream>>>(D2, 0.0f,    N_NODES);
    fill_f32<<<cdiv(N_NODES * DIM_OUT,256),256, 0, stream>>>(O2, 0.0f,    N_NODES * DIM_OUT);

    // --- layer 1: linear projections (WMMA fp32) ---
    {
        int tiles = (N_NODES / 16) * (H1 / 16);               // 625*16 = 10000
        gemm_wmma_f32<<<cdiv(tiles, 8), 256, 0, stream>>>(x, Wl1, bl1, XL1, N_NODES, DIM_IN, H1);
        gemm_wmma_f32<<<cdiv(tiles, 8), 256, 0, stream>>>(x, Wr1, br1, XR1, N_NODES, DIM_IN, H1);
    }

    // --- layer 1: edge passes ---
    edge_logits1<<<cdiv(N_EDGES * HEADS, 256), 256, 0, stream>>>(XL1, XR1, src, dst, att1, LG1, M1);
    edge_exp1   <<<cdiv(N_EDGES * HEADS, 256), 256, 0, stream>>>(dst, LG1, M1, D1);
    edge_msg1   <<<N_EDGES, 256, 0, stream>>>(XL1, src, dst, LG1, D1, O1);
    elu_bias    <<<cdiv(N_NODES * H1, 256), 256, 0, stream>>>(O1, bias1, N_NODES * H1);

    // --- layer 2: linear projections (WMMA fp32) ---
    {
        int tiles = (N_NODES / 16) * (DIM_OUT / 16);          // 625*4 = 2500
        gemm_wmma_f32<<<cdiv(tiles, 8), 256, 0, stream>>>(O1, Wl2, bl2, XL2, N_NODES, H1, DIM_OUT);
        gemm_wmma_f32<<<cdiv(tiles, 8), 256, 0, stream>>>(O1, Wr2, br2, XR2, N_NODES, H1, DIM_OUT);
    }

    // --- layer 2: edge passes (1 head) ---
    edge_logits2<<<cdiv(N_EDGES, 256), 256, 0, stream>>>(XL2, XR2, src, dst, att2, LG2, M2);
    edge_exp2   <<<cdiv(N_EDGES, 256), 256, 0, stream>>>(dst, LG2, M2, D2);
    edge_msg2   <<<cdiv(N_EDGES * DIM_OUT, 256), 256, 0, stream>>>(XL2, src, dst, LG2, D2, O2);

    // --- final log-softmax (+bias2) straight into d_out ---
    log_softmax64<<<cdiv(N_NODES, 8), 256, 0, stream>>>(O2, bias2, (float*)d_out, N_NODES);
}